// EGRETLayer_88519275970734
// MI455X (gfx1250) — compile-verified
//
#include <hip/hip_runtime.h>
#include <hip/hip_bf16.h>
#include <math.h>

#define IN_DIM  128
#define OUT_DIM 64
#define ED_DIM  32

// Swizzled B-staging: W[k][n], W[k+1][n] stored adjacently so one ds_load_b64
// fetches a whole B fragment. Row-pair stride 160 dwords == 32 (mod 64) banks,
// so lanes 0-15 (K=k0,k0+1) and lanes 16-31 (K=k0+2,k0+3) hit disjoint bank sets.
#define BSTRIDE_V2 80                      // stride per k-pair row, in v2f units
#define BSTRIDE_F  (2 * BSTRIDE_V2)        // ... in floats

typedef __attribute__((ext_vector_type(2))) float v2f;
typedef __attribute__((ext_vector_type(8))) float v8f;

// ---------------------------------------------------------------------------
// f32 WMMA helper: one wave computes a [16 x 64] tile = A[16 x K] * W[K x 64]
// via V_WMMA_F32_16X16X4_F32 (wave32, full fp32 — matches fp32 reference).
// A 16x4 layout (ISA 7.12.2): lanes 0-15 K={k0,k0+1}, lanes 16-31 K={k0+2,k0+3},
// M = lane&15.  B mirrors with N = lane&15.  C/D: VGPR r -> M = r + 8*(lane>>4).
// ---------------------------------------------------------------------------
struct Acc4 { v8f t[4]; };

__device__ __attribute__((always_inline)) inline
Acc4 wave_gemm_tile(const float* __restrict__ arow,   // lane's A row (8B aligned)
                    const float* __restrict__ lWs,    // swizzled W in LDS
                    int kdim, int half, int l16)
{
    Acc4 A;
    A.t[0] = {}; A.t[1] = {}; A.t[2] = {}; A.t[3] = {};
    const v2f* a2 = (const v2f*)arow;
    for (int k0 = 0; k0 < kdim; k0 += 4) {
        const int kp  = (k0 >> 1) + half;             // k-pair index for this lane
        const v2f  a  = a2[kp];                       // {A[m][kk], A[m][kk+1]}
        const v2f* br = (const v2f*)(lWs + kp * BSTRIDE_F);
#pragma unroll
        for (int nt = 0; nt < 4; ++nt) {
            const v2f b = br[nt * 16 + l16];          // one ds_load_b64, no movs
            A.t[nt] = __builtin_amdgcn_wmma_f32_16x16x4_f32(
                false, a, false, b, (short)0, A.t[nt], false, false);
        }
    }
    return A;
}

// Stage W[kdim x 64] (row-major) into swizzled LDS layout (block-cooperative).
__device__ __attribute__((always_inline)) inline
void stage_w(float* __restrict__ lWs, const float* __restrict__ W, int kdim)
{
    const int total = kdim * OUT_DIM;
    for (int i = threadIdx.x; i < total; i += blockDim.x) {
        const int k = i >> 6;           // / OUT_DIM
        const int n = i & 63;           // % OUT_DIM
        lWs[(k >> 1) * BSTRIDE_F + (n << 1) + (k & 1)] = W[i];
    }
}

// float atomic max via sign-magnitude trick (target init must be -inf)
__device__ inline void atomicMaxF(float* addr, float val) {
    if (val >= 0.0f) atomicMax(reinterpret_cast<int*>(addr), __float_as_int(val));
    else             atomicMin(reinterpret_cast<unsigned int*>(addr), __float_as_uint(val));
}

// ---------------------------------------------------------------------------
// Kernel 1: z = x@W_fc + b_fc ; h_init = x@W_self + b_self
// 256 threads = 8 waves; each wave owns one 16-node tile. 40KB swizzled LDS,
// reused across both GEMM phases.
// ---------------------------------------------------------------------------
__global__ void k_node_gemm(const float* __restrict__ x,
                            const float* __restrict__ W_fc,  const float* __restrict__ b_fc,
                            const float* __restrict__ W_self, const float* __restrict__ b_self,
                            float* __restrict__ z, float* __restrict__ hout,
                            int N, int nTiles)
{
    __shared__ float lWs[(IN_DIM / 2) * BSTRIDE_F];   // 40 KB
    const int lane = threadIdx.x & 31;
    const int wave = threadIdx.x >> 5;
    const int half = lane >> 4;
    const int l16  = lane & 15;
    const int tile = blockIdx.x * 8 + wave;

    int mrow = tile * 16 + l16;
    if (mrow >= N) mrow = N - 1;                // clamp: EXEC stays full for WMMA
    const float* xrow = x + (size_t)mrow * IN_DIM;

    // ---- phase 1: z = x @ W_fc + b_fc ----
    stage_w(lWs, W_fc, IN_DIM);
    __syncthreads();
    if (tile < nTiles) {
        Acc4 acc = wave_gemm_tile(xrow, lWs, IN_DIM, half, l16);
#pragma unroll
        for (int nt = 0; nt < 4; ++nt) {
            const int   c  = nt * 16 + l16;
            const float bc = b_fc[c];
#pragma unroll
            for (int r = 0; r < 8; ++r) {
                const int row = tile * 16 + r + 8 * half;
                if (row < N) z[(size_t)row * OUT_DIM + c] = acc.t[nt][r] + bc;
            }
        }
    }
    __syncthreads();

    // ---- phase 2: h_init = x @ W_self + b_self ----
    stage_w(lWs, W_self, IN_DIM);
    __syncthreads();
    if (tile < nTiles) {
        Acc4 acc = wave_gemm_tile(xrow, lWs, IN_DIM, half, l16);
#pragma unroll
        for (int nt = 0; nt < 4; ++nt) {
            const int   c  = nt * 16 + l16;
            const float bc = b_self[c];
#pragma unroll
            for (int r = 0; r < 8; ++r) {
                const int row = tile * 16 + r + 8 * half;
                if (row < N) hout[(size_t)row * OUT_DIM + c] = acc.t[nt][r] + bc;
            }
        }
    }
}

// ---------------------------------------------------------------------------
// Kernel 2: per-node attention scalars a_src = z.Wa[0:64], a_dst = z.Wa[64:128];
// init segment max to -inf, denom to 0.
// ---------------------------------------------------------------------------
__global__ void k_node_attn(const float* __restrict__ z, const float* __restrict__ W_attn,
                            float* __restrict__ a_src, float* __restrict__ a_dst,
                            float* __restrict__ m, float* __restrict__ denom, int N)
{
    const int n = blockIdx.x * blockDim.x + threadIdx.x;
    if (n >= N) return;
    const float* zr = z + (size_t)n * OUT_DIM;
    float as = 0.f, ad = 0.f;
#pragma unroll 8
    for (int c = 0; c < OUT_DIM; ++c) {
        const float zc = zr[c];
        as += zc * W_attn[c];
        ad += zc * W_attn[OUT_DIM + c];
    }
    a_src[n] = as;
    a_dst[n] = ad;
    m[n]     = -INFINITY;
    denom[n] = 0.0f;
}

// ---------------------------------------------------------------------------
// Kernel 3: edge logits e = leaky_relu(a_src[row]+a_dst[col]+ea.Wa[128:160]+b)
// plus float atomic segment-max into m[col].
// ---------------------------------------------------------------------------
__global__ void k_edge_logit(const float* __restrict__ edge_attr,
                             const int* __restrict__ erow, const int* __restrict__ ecol,
                             const float* __restrict__ W_attn, const float* __restrict__ b_attn,
                             const float* __restrict__ a_src, const float* __restrict__ a_dst,
                             float* __restrict__ elog, float* __restrict__ m, int E)
{
    const int e = blockIdx.x * blockDim.x + threadIdx.x;
    if (e >= E) return;
    const float* ea = edge_attr + (size_t)e * ED_DIM;
    float dot = 0.f;
#pragma unroll
    for (int k = 0; k < ED_DIM; ++k) dot += ea[k] * W_attn[2 * OUT_DIM + k];
    const int c = ecol[e];
    const float s  = a_src[erow[e]] + a_dst[c] + dot + b_attn[0];
    const float lr = (s > 0.0f) ? s : 0.2f * s;   // leaky_relu(0.2)
    elog[e] = lr;
    atomicMaxF(&m[c], lr);
}

// ---------------------------------------------------------------------------
// Kernel 4: p = exp(e - m[col]); denom[col] += p
// ---------------------------------------------------------------------------
__global__ void k_edge_soft(const float* __restrict__ elog, const int* __restrict__ ecol,
                            const float* __restrict__ m, float* __restrict__ pbuf,
                            float* __restrict__ denom, int E)
{
    const int e = blockIdx.x * blockDim.x + threadIdx.x;
    if (e >= E) return;
    const int c = ecol[e];
    const float p = expf(elog[e] - m[c]);
    pbuf[e] = p;
    atomicAdd(&denom[c], p);
}

// ---------------------------------------------------------------------------
// Kernel 5: ez = edge_attr @ W_edge (WMMA, 16-edge tiles, K=32) fused with
// h[col] += alpha * (z[row] + ez + b_edge).  64 f32 atomic adds per edge.
// ---------------------------------------------------------------------------
__global__ void k_edge_aggregate(const float* __restrict__ edge_attr,
                                 const int* __restrict__ erow, const int* __restrict__ ecol,
                                 const float* __restrict__ W_edge, const float* __restrict__ b_edge,
                                 const float* __restrict__ z, const float* __restrict__ pbuf,
                                 const float* __restrict__ denom, float* __restrict__ hout,
                                 int E, int nTiles)
{
    __shared__ float lWs[(ED_DIM / 2) * BSTRIDE_F];   // 10 KB
    const int lane = threadIdx.x & 31;
    const int wave = threadIdx.x >> 5;
    const int half = lane >> 4;
    const int l16  = lane & 15;
    const int tile = blockIdx.x * 8 + wave;

    stage_w(lWs, W_edge, ED_DIM);
    __syncthreads();
    if (tile >= nTiles) return;                 // wave-uniform; no barriers below

    int arow = tile * 16 + l16;
    if (arow >= E) arow = E - 1;                // clamp: EXEC stays full for WMMA
    const float* ea = edge_attr + (size_t)arow * ED_DIM;

    // Prefetch next tile's edge_attr row for this lane (streaming 102MB input).
    {
        const size_t nxt = (size_t)arow + 16 * 8;     // next tile handled by this wave slot
        if (nxt < (size_t)E)
            __builtin_prefetch(edge_attr + nxt * ED_DIM, 0, 1);
    }

    Acc4 acc = wave_gemm_tile(ea, lWs, ED_DIM, half, l16);   // ez tile [16 x 64]

    // Per-output-row edge metadata (row index within tile = r + 8*half)
    float alph[8];
    int   cr[8], rr[8];
#pragma unroll
    for (int r = 0; r < 8; ++r) {
        const int eidx = tile * 16 + r + 8 * half;
        if (eidx < E) {
            const int c = ecol[eidx];
            cr[r]   = c;
            rr[r]   = erow[eidx];
            alph[r] = pbuf[eidx] / fmaxf(denom[c], 1e-16f);
        } else {
            cr[r] = -1; rr[r] = 0; alph[r] = 0.f;
        }
    }

#pragma unroll
    for (int nt = 0; nt < 4; ++nt) {
        const int   c  = nt * 16 + l16;
        const float bc = b_edge[c];
#pragma unroll
        for (int r = 0; r < 8; ++r) {
            if (cr[r] >= 0) {
                const float val = alph[r] * (z[(size_t)rr[r] * OUT_DIM + c] + acc.t[nt][r] + bc);
                atomicAdd(&hout[(size_t)cr[r] * OUT_DIM + c], val);
            }
        }
    }
}

// ---------------------------------------------------------------------------
extern "C" void kernel_launch(void* const* d_in, const int* in_sizes, int n_in,
                              void* d_out, int out_size, void* d_ws, size_t ws_size,
                              hipStream_t stream)
{
    const float* x         = (const float*)d_in[0];
    const float* edge_attr = (const float*)d_in[1];
    const int*   edge_idx  = (const int*)  d_in[2];
    const float* W_fc      = (const float*)d_in[3];
    const float* b_fc      = (const float*)d_in[4];
    const float* W_attn    = (const float*)d_in[5];
    const float* b_attn    = (const float*)d_in[6];
    const float* W_edge    = (const float*)d_in[7];
    const float* b_edge    = (const float*)d_in[8];
    const float* W_self    = (const float*)d_in[9];
    const float* b_self    = (const float*)d_in[10];

    const int N = in_sizes[0] / IN_DIM;
    const int E = in_sizes[1] / ED_DIM;
    const int* erow = edge_idx;         // edge_index[0] (src)
    const int* ecol = edge_idx + E;     // edge_index[1] (dst)

    // workspace (floats): z[N*64] | a_src[N] | a_dst[N] | m[N] | denom[N] | elog[E] | p[E]
    float* ws     = (float*)d_ws;
    float* z      = ws;
    float* a_src  = z + (size_t)N * OUT_DIM;
    float* a_dst  = a_src + N;
    float* m      = a_dst + N;
    float* denom  = m + N;
    float* elog   = denom + N;
    float* pbuf   = elog + E;
    float* hout   = (float*)d_out;

    const int nNodeTiles = (N + 15) / 16;
    const int nEdgeTiles = (E + 15) / 16;

    k_node_gemm<<<(nNodeTiles + 7) / 8, 256, 0, stream>>>(
        x, W_fc, b_fc, W_self, b_self, z, hout, N, nNodeTiles);

    k_node_attn<<<(N + 255) / 256, 256, 0, stream>>>(
        z, W_attn, a_src, a_dst, m, denom, N);

    k_edge_logit<<<(E + 255) / 256, 256, 0, stream>>>(
        edge_attr, erow, ecol, W_attn, b_attn, a_src, a_dst, elog, m, E);

    k_edge_soft<<<(E + 255) / 256, 256, 0, stream>>>(
        elog, ecol, m, pbuf, denom, E);

    k_edge_aggregate<<<(nEdgeTiles + 7) / 8, 256, 0, stream>>>(
        edge_attr, erow, ecol, W_edge, b_edge, z, pbuf, denom, hout, E, nEdgeTiles);
}